// RetinaNetLoss_31052613550332
// MI455X (gfx1250) — compile-verified
//
#include <hip/hip_runtime.h>
#include <hip/hip_bf16.h>

#define NCLASSES 21
#define MAX_T 512

typedef __attribute__((ext_vector_type(2))) float v2f;
typedef __attribute__((ext_vector_type(8))) float v8f;

__device__ __forceinline__ float smooth_l1(float d) {
    d = fabsf(d);
    return d < 1.0f ? 0.5f * d * d : d - 0.5f;
}

__global__ void retina_init_ws(float* ws) {
    ws[0] = 0.0f;  // cls_sum
    ws[1] = 0.0f;  // reg_sum
    ws[2] = 0.0f;  // pos_cnt
}

__global__ __launch_bounds__(256) void retina_main(
    const float* __restrict__ cls,        // [A, 21]
    const float4* __restrict__ boxp,      // [A]
    const float4* __restrict__ anc,       // [A]
    const float4* __restrict__ tboxes,    // [T]
    const int* __restrict__ tlabels,      // [T]
    float* __restrict__ ws,
    int A, int T)
{
    __shared__ float tx1[MAX_T], ty1[MAX_T], tx2[MAX_T], ty2[MAX_T], tarea[MAX_T];
    __shared__ int   tlab[MAX_T];
    __shared__ float sCls, sReg, sPos;

    if (threadIdx.x == 0) { sCls = 0.0f; sReg = 0.0f; sPos = 0.0f; }
    for (int t = threadIdx.x; t < T; t += blockDim.x) {
        float4 b = tboxes[t];
        tx1[t] = b.x; ty1[t] = b.y; tx2[t] = b.z; ty2[t] = b.w;
        tarea[t] = (b.z - b.x) * (b.w - b.y);
        tlab[t] = tlabels[t];
    }
    __syncthreads();

    int gi = blockIdx.x * blockDim.x + threadIdx.x;
    bool valid = gi < A;
    int i = valid ? gi : 0;   // clamp instead of early-exit: keep EXEC full for WMMA

    float4 a = anc[i];
    float areaA = (a.z - a.x) * (a.w - a.y);

    // --- max IoU + argmax over T targets (LDS broadcast reads) ---
    float best = -1.0f;
    int tid = 0;
    #pragma unroll 4
    for (int t = 0; t < T; ++t) {
        float lx = fmaxf(tx1[t], a.x), ly = fmaxf(ty1[t], a.y);
        float rx = fminf(tx2[t], a.z), ry = fminf(ty2[t], a.w);
        float w = fmaxf(rx - lx, 0.0f), h = fmaxf(ry - ly, 0.0f);
        float inter = w * h;
        float uni = fmaxf(tarea[t] + areaA - inter, 1e-8f);
        float iou = inter / uni;
        if (iou > best) { best = iou; tid = t; }   // first-max, matches jnp.argmax
    }
    bool pos = best >= 0.5f;
    int tgtc = pos ? tlab[tid] : 0;   // background = class 0 one-hot

    // --- focal loss over 21 classes ---
    float clsAcc = 0.0f;
    const float* crow = cls + (size_t)i * NCLASSES;
    #pragma unroll
    for (int c = 0; c < NCLASSES; ++c) {
        float x = crow[c];
        float p = 1.0f / (1.0f + expf(-x));
        float l1p = log1pf(expf(-fabsf(x)));          // shared term of logsigmoid(+/-x)
        bool ist = (c == tgtc);
        float ls = (ist ? fminf(x, 0.0f) : fminf(-x, 0.0f)) - l1p;
        float ce = -ls;
        float pt = ist ? p : (1.0f - p);
        float at = ist ? 0.25f : 0.75f;               // ALPHA / (1-ALPHA)
        float om = 1.0f - pt;
        clsAcc += at * om * om * ce;                  // GAMMA = 2 -> square
    }

    // --- regression: encode + smooth L1, gated by pos ---
    float regAcc = 0.0f;
    {
        float4 bp = boxp[i];
        float aw = a.z - a.x, ah = a.w - a.y;
        float acx = (a.x + a.z) * 0.5f, acy = (a.y + a.w) * 0.5f;
        float e0 = ((bp.x + bp.z) * 0.5f - acx) / aw;
        float e1 = ((bp.y + bp.w) * 0.5f - acy) / ah;
        float e2 = logf((bp.z - bp.x) / aw);
        float e3 = logf((bp.w - bp.y) / ah);
        float t0 = ((tx1[tid] + tx2[tid]) * 0.5f - acx) / aw;
        float t1 = ((ty1[tid] + ty2[tid]) * 0.5f - acy) / ah;
        float t2 = logf((tx2[tid] - tx1[tid]) / aw);
        float t3 = logf((ty2[tid] - ty1[tid]) / ah);
        float r = smooth_l1(e0 - t0) + smooth_l1(e1 - t1) +
                  smooth_l1(e2 - t2) + smooth_l1(e3 - t3);
        regAcc = pos ? r : 0.0f;
    }
    if (!valid) { clsAcc = 0.0f; regAcc = 0.0f; pos = false; }
    float posf = pos ? 1.0f : 0.0f;

    // --- wave32 reduction via V_WMMA_F32_16X16X4_F32 ---
    // A: VGPR0 = cls partial (K=0 lanes 0-15 / K=2 lanes 16-31),
    //    VGPR1 = reg partial (K=1 / K=3).
    // B column 0 sums the cls slots (B[0,0]=lane0.v0, B[2,0]=lane16.v0),
    // B column 1 sums the reg slots (B[1,1]=lane1.v1, B[3,1]=lane17.v1).
    int lane = threadIdx.x & 31;
    v2f Av; Av.x = clsAcc; Av.y = regAcc;
    v2f Bv;
    Bv.x = (lane == 0 || lane == 16) ? 1.0f : 0.0f;
    Bv.y = (lane == 1 || lane == 17) ? 1.0f : 0.0f;
    v8f Dm = {};
    Dm = __builtin_amdgcn_wmma_f32_16x16x4_f32(false, Av, false, Bv,
                                               (short)0, Dm, false, false);
    float s = Dm[0] + Dm[1] + Dm[2] + Dm[3] + Dm[4] + Dm[5] + Dm[6] + Dm[7];
    // lane 0:  rows 0-7  of col 0 (cls),  lane 16: rows 8-15 of col 0 (cls)
    // lane 1:  rows 0-7  of col 1 (reg),  lane 17: rows 8-15 of col 1 (reg)

    v2f Ap; Ap.x = posf; Ap.y = 0.0f;
    v8f Dp = {};
    Dp = __builtin_amdgcn_wmma_f32_16x16x4_f32(false, Ap, false, Bv,
                                               (short)0, Dp, false, false);
    float sp = Dp[0] + Dp[1] + Dp[2] + Dp[3] + Dp[4] + Dp[5] + Dp[6] + Dp[7];

    if (lane == 0 || lane == 16) { atomicAdd(&sCls, s); atomicAdd(&sPos, sp); }
    if (lane == 1 || lane == 17) { atomicAdd(&sReg, s); }
    __syncthreads();

    if (threadIdx.x == 0) {
        atomicAdd(&ws[0], sCls);
        atomicAdd(&ws[1], sReg);
        atomicAdd(&ws[2], sPos);
    }
}

__global__ void retina_finalize(const float* __restrict__ ws, float* __restrict__ out) {
    float cnt = ws[2];
    float norm = fmaxf(cnt, 1.0f);
    float wc = ws[0] / norm;                               // CLS_W = 1
    float wr = (cnt > 0.0f) ? ws[1] / (norm * 4.0f) : 0.0f; // REG_W = 1
    out[0] = wc + wr;
    out[1] = wc;
    out[2] = wr;
}

extern "C" void kernel_launch(void* const* d_in, const int* in_sizes, int n_in,
                              void* d_out, int out_size, void* d_ws, size_t ws_size,
                              hipStream_t stream) {
    const float*  cls     = (const float*)d_in[0];   // [B,A,21]
    const float4* boxp    = (const float4*)d_in[1];  // [B,A,4]
    const float4* anc     = (const float4*)d_in[2];  // [B,A,4]
    const float4* tboxes  = (const float4*)d_in[3];  // [B,T,4]
    const int*    tlabels = (const int*)d_in[4];     // [B,T]
    float* out = (float*)d_out;
    float* ws  = (float*)d_ws;

    int A = in_sizes[2] / 4;   // total anchors (batch flattened) = 196608
    int T = in_sizes[4];       // total targets (batch flattened) = 200
    if (T > MAX_T) T = MAX_T;  // defensive; reference T = 200

    retina_init_ws<<<1, 1, 0, stream>>>(ws);
    int blocks = (A + 255) / 256;
    retina_main<<<blocks, 256, 0, stream>>>(cls, boxp, anc, tboxes, tlabels, ws, A, T);
    retina_finalize<<<1, 1, 0, stream>>>(ws, out);
}